// HANLayer_60318520705163
// MI455X (gfx1250) — compile-verified
//
#include <hip/hip_runtime.h>
#include <math.h>

typedef float v2f __attribute__((ext_vector_type(2)));
typedef float v8f __attribute__((ext_vector_type(8)));

constexpr int   NN   = 50000;    // nodes
constexpr long  NE   = 800000;   // edges per metapath
constexpr int   P    = 2;        // metapaths
constexpr int   H    = 4;        // heads
constexpr int   F    = 32;       // out per head
constexpr int   HF   = 128;      // H*F == IN == HID
constexpr float SLOPE = 0.2f;

#define LDSB_STRIDE 136   // 128 + 8 pad: laneHi rows land 16 banks away -> conflict-free
#define LDSA_STRIDE 132   // 128 + 4 pad: bank = (4*row + k) % 64 -> conflict-free

// ---------------------------------------------------------------------------
// C[rowTile 16 x 128] = A[16 x 128] @ W[128 x 128]  via V_WMMA_F32_16X16X4_F32
// 256 threads = 8 waves; wave w computes N-columns [16w, 16w+16).
// act==1: fused epilogue v = tanh(v + bias[col])
// ---------------------------------------------------------------------------
__global__ __launch_bounds__(256) void gemm128_wmma(
    const float* __restrict__ A, long aStride,
    const float* __restrict__ W, long wStride,
    const float* __restrict__ bias,
    float* __restrict__ Out, long outStride, int act)
{
  __shared__ float ldsB[128 * LDSB_STRIDE];
  __shared__ float ldsA[16 * LDSA_STRIDE];
  const int tid = threadIdx.x;
  const float* Wp = W + (long)blockIdx.y * wStride;
  const float* Ap = A + (long)blockIdx.y * aStride;
  float*       Op = Out + (long)blockIdx.y * outStride;
  const long rowBase = (long)blockIdx.x * 16;

  // cooperative load: weights 128x128 (coalesced float4; rows are 4-aligned)
#pragma unroll
  for (int i = 0; i < 16; ++i) {
    int j = (tid + i * 256) * 4;
    int r = j >> 7, c = j & 127;
    *(float4*)&ldsB[r * LDSB_STRIDE + c] = *(const float4*)&Wp[j];
  }
  // cooperative load: A tile 16x128 (rows contiguous in memory -> coalesced)
#pragma unroll
  for (int i = 0; i < 2; ++i) {
    int j = (tid + i * 256) * 4;
    int r = j >> 7, c = j & 127;
    *(float4*)&ldsA[r * LDSA_STRIDE + c] = *(const float4*)&Ap[rowBase * 128 + j];
  }
  __syncthreads();

  const int wave    = tid >> 5;
  const int lane    = tid & 31;
  const int laneLo  = lane & 15;
  const int laneHi  = lane >> 4;
  const int colBase = wave * 16;

  v8f acc = {};
#pragma unroll
  for (int k0 = 0; k0 < 128; k0 += 4) {
    const int k = k0 + 2 * laneHi;           // ISA 7.12.2: A vgpr v holds K=k+v
    v2f a, b;
    a.x = ldsA[laneLo * LDSA_STRIDE + k];
    a.y = ldsA[laneLo * LDSA_STRIDE + k + 1];
    b.x = ldsB[k * LDSB_STRIDE + colBase + laneLo];        // B vgpr v holds row K=k+v
    b.y = ldsB[(k + 1) * LDSB_STRIDE + colBase + laneLo];
    acc = __builtin_amdgcn_wmma_f32_16x16x4_f32(false, a, false, b,
                                                (short)0, acc, false, false);
  }

  const int col = colBase + laneLo;
  const float bv = bias ? bias[col] : 0.0f;
#pragma unroll
  for (int j = 0; j < 8; ++j) {              // D vgpr j: M = j + 8*laneHi, N = laneLo
    long row = rowBase + j + 8 * laneHi;
    float v = acc[j];
    if (act == 1) v = tanhf(v + bv);
    Op[row * 128 + col] = v;
  }
}

// ---------------------------------------------------------------------------
// el/er: per (p, n, h) dot of feat[n,h,:] with attn vectors
// ---------------------------------------------------------------------------
__global__ void scores_kernel(const float* __restrict__ feat,
                              const float* __restrict__ attn_l,
                              const float* __restrict__ attn_r,
                              float* __restrict__ el, float* __restrict__ er)
{
  int idx = blockIdx.x * blockDim.x + threadIdx.x;          // [p][n][h]
  if (idx >= P * NN * H) return;
  int h = idx & 3;
  int n = (idx >> 2) % NN;
  int p = idx / (NN * H);
  const float* f  = feat + ((long)p * NN + n) * 128 + h * F;
  const float* al = attn_l + (p * H + h) * F;
  const float* ar = attn_r + (p * H + h) * F;
  float sl = 0.f, sr = 0.f;
#pragma unroll
  for (int i = 0; i < F; ++i) { float v = f[i]; sl += v * al[i]; sr += v * ar[i]; }
  el[idx] = sl; er[idx] = sr;
}

// monotonic float <-> uint mapping for atomicMax
__device__ __forceinline__ unsigned fmap(float f) {
  unsigned u = __float_as_uint(f);
  return (u & 0x80000000u) ? ~u : (u | 0x80000000u);
}
__device__ __forceinline__ float funmap(unsigned u) {
  unsigned b = (u & 0x80000000u) ? (u ^ 0x80000000u) : ~u;
  return __uint_as_float(b);
}

__global__ void edge_pass1(const int* __restrict__ src, const int* __restrict__ dst,
                           const float* __restrict__ el, const float* __restrict__ er,
                           float* __restrict__ ebuf, unsigned* __restrict__ maxb)
{
  long e = (long)blockIdx.x * blockDim.x + threadIdx.x;
  if (e >= NE) return;
  int s = src[e], d = dst[e];
#pragma unroll
  for (int h = 0; h < H; ++h) {
    float v = el[s * H + h] + er[d * H + h];
    v = v > 0.f ? v : SLOPE * v;                           // leaky_relu
    ebuf[e * H + h] = v;
    atomicMax(&maxb[d * H + h], fmap(v));
  }
}

__global__ void edge_pass2(const int* __restrict__ dst, float* __restrict__ ebuf,
                           const unsigned* __restrict__ maxb, float* __restrict__ den)
{
  long e = (long)blockIdx.x * blockDim.x + threadIdx.x;
  if (e >= NE) return;
  int d = dst[e];
#pragma unroll
  for (int h = 0; h < H; ++h) {
    unsigned raw = maxb[d * H + h];
    float m = raw ? funmap(raw) : 0.0f;                    // 0 for zero-in-degree
    float ex = expf(ebuf[e * H + h] - m);
    ebuf[e * H + h] = ex;
    atomicAdd(&den[d * H + h], ex);
  }
}

// one thread per (edge, column): coalesced gather + coalesced atomic scatter
__global__ __launch_bounds__(256) void edge_pass3(
    const int* __restrict__ src, const int* __restrict__ dst,
    const float* __restrict__ ebuf, const float* __restrict__ den,
    const float* __restrict__ feat, float* __restrict__ out)
{
  long gid = (long)blockIdx.x * 256 + threadIdx.x;          // NE*128 exactly
  int col  = (int)(gid & 127);
  long e   = gid >> 7;
  if (e >= NE) return;
  int s = src[e], d = dst[e];
  int h = col >> 5;
  float alpha = ebuf[e * H + h] / fmaxf(den[d * H + h], 1e-9f);
  atomicAdd(&out[(long)d * 128 + col], feat[(long)s * 128 + col] * alpha);
}

__global__ void z_finalize(float* __restrict__ z, const float* __restrict__ bias)
{
  long idx = (long)blockIdx.x * blockDim.x + threadIdx.x;   // P*NN*128
  if (idx >= (long)P * NN * 128) return;
  int col = (int)(idx & 127);
  int p   = (int)(idx / ((long)NN * 128));
  float x = z[idx] + bias[p * 128 + col];
  z[idx] = x > 0.f ? x : expm1f(x);                         // elu
}

__global__ void sem_score(const float* __restrict__ proj,
                          const float* __restrict__ w2, float* __restrict__ wsum)
{
  long row = (long)blockIdx.x * blockDim.x + threadIdx.x;   // P*NN rows
  if (row >= (long)P * NN) return;
  const float* pr = proj + row * 128;
  float s = 0.f;
#pragma unroll 8
  for (int i = 0; i < 128; ++i) s += pr[i] * w2[i];
  atomicAdd(&wsum[row / NN], s);
}

__global__ void beta_kernel(const float* __restrict__ wsum, float* __restrict__ beta)
{
  float w0 = wsum[0] / (float)NN, w1 = wsum[1] / (float)NN;
  float m = fmaxf(w0, w1);
  float e0 = expf(w0 - m), e1 = expf(w1 - m);
  float s = e0 + e1;
  beta[0] = e0 / s; beta[1] = e1 / s;
}

__global__ void final_combine(const float* __restrict__ z,
                              const float* __restrict__ beta, float* __restrict__ out)
{
  long idx = (long)blockIdx.x * blockDim.x + threadIdx.x;   // NN*128
  if (idx >= (long)NN * 128) return;
  out[idx] = beta[0] * z[idx] + beta[1] * z[(long)NN * 128 + idx];
}

// ---------------------------------------------------------------------------
extern "C" void kernel_launch(void* const* d_in, const int* in_sizes, int n_in,
                              void* d_out, int out_size, void* d_ws, size_t ws_size,
                              hipStream_t stream)
{
  const float* h      = (const float*)d_in[0];
  const int*   src    = (const int*)  d_in[1];
  const int*   dst    = (const int*)  d_in[2];
  const float* fc_w   = (const float*)d_in[3];
  const float* attn_l = (const float*)d_in[4];
  const float* attn_r = (const float*)d_in[5];
  const float* bias   = (const float*)d_in[6];
  const float* sem_w1 = (const float*)d_in[7];
  const float* sem_b1 = (const float*)d_in[8];
  const float* sem_w2 = (const float*)d_in[9];
  float* out = (float*)d_out;

  float* ws = (float*)d_ws;
  long off = 0;
  float* feat = ws + off; off += (long)P * NN * 128;        // also reused for proj
  float* z    = ws + off; off += (long)P * NN * 128;
  float* ebuf = ws + off; off += NE * H;
  float* el   = ws + off; off += (long)P * NN * H;
  float* er   = ws + off; off += (long)P * NN * H;
  unsigned* maxb = (unsigned*)(ws + off); off += (long)NN * H;
  float* den  = ws + off; off += (long)NN * H;
  float* wsum = ws + off; off += 2;
  float* beta = ws + off; off += 2;

  // 1) feat[p] = h @ fc_w[p]  (WMMA f32 16x16x4, K=128)
  gemm128_wmma<<<dim3(NN / 16, P), 256, 0, stream>>>(
      h, 0L, fc_w, (long)128 * 128, nullptr, feat, (long)NN * 128, 0);

  // 2) node attention scores
  scores_kernel<<<(P * NN * H + 255) / 256, 256, 0, stream>>>(feat, attn_l, attn_r, el, er);

  // 3) per-metapath edge softmax + aggregation
  hipMemsetAsync(z, 0, (size_t)P * NN * 128 * sizeof(float), stream);
  for (int p = 0; p < P; ++p) {
    hipMemsetAsync(maxb, 0, (size_t)NN * H * sizeof(unsigned), stream);
    hipMemsetAsync(den,  0, (size_t)NN * H * sizeof(float), stream);
    const int*   sp  = src + (long)p * NE;
    const int*   dp  = dst + (long)p * NE;
    const float* elp = el + (long)p * NN * H;
    const float* erp = er + (long)p * NN * H;
    const float* fp  = feat + (long)p * NN * 128;
    float*       zp  = z + (long)p * NN * 128;
    edge_pass1<<<(int)((NE + 255) / 256), 256, 0, stream>>>(sp, dp, elp, erp, ebuf, maxb);
    edge_pass2<<<(int)((NE + 255) / 256), 256, 0, stream>>>(dp, ebuf, maxb, den);
    edge_pass3<<<(int)(NE * 128 / 256), 256, 0, stream>>>(sp, dp, ebuf, den, fp, zp);
  }

  // 4) z = elu(out + bias)
  z_finalize<<<(int)(((long)P * NN * 128 + 255) / 256), 256, 0, stream>>>(z, bias);

  // 5) semantic attention: proj = tanh(z @ sem_w1 + b1) (WMMA, fused epilogue)
  gemm128_wmma<<<dim3(P * NN / 16, 1), 256, 0, stream>>>(
      z, 0L, sem_w1, 0L, sem_b1, feat, 0L, 1);

  hipMemsetAsync(wsum, 0, 2 * sizeof(float), stream);
  sem_score<<<(P * NN + 255) / 256, 256, 0, stream>>>(feat, sem_w2, wsum);
  beta_kernel<<<1, 1, 0, stream>>>(wsum, beta);

  // 6) out = sum_p beta[p] * z[:,p,:]
  final_combine<<<(int)(((long)NN * 128 + 255) / 256), 256, 0, stream>>>(z, beta, out);
}